// StockS4_3066606650069
// MI455X (gfx1250) — compile-verified
//
#include <hip/hip_runtime.h>
#include <math.h>

#define BB 32
#define LL 2048
#define HH 256
#define NN 32
#define NLAYER 4
#define DOUTN 200
#define TT 64
#define CC (LL / TT)
#define LN_EPS 1e-5f

typedef float v2f __attribute__((ext_vector_type(2)));
typedef float v8f __attribute__((ext_vector_type(8)));

// ---------------- encoder: h[b,h,l] = x[b,l]*enc_w[h] + enc_b[h] ----------------
__global__ void enc_kernel(const float* __restrict__ x, const float* __restrict__ enc_w,
                           const float* __restrict__ enc_b, float* __restrict__ hbuf) {
  size_t idx = (size_t)blockIdx.x * blockDim.x + threadIdx.x;
  if (idx >= (size_t)BB * HH * LL) return;
  int l = (int)(idx % LL);
  size_t t = idx / LL;
  int h = (int)(t % HH);
  int b = (int)(t / HH);
  hbuf[idx] = x[(size_t)b * LL + l] * enc_w[h] + enc_b[h];
}

// ---------------- per-layer S4D coefficient precompute ----------------
// grid = H blocks, 32 lanes (one per mode n). Computes:
//   ktap[h][m]      = 2 Re(Ct w^m)              (truncated conv kernel, m<T)
//   Er/Ei[h][t][n]  = 2 Re/Im(Ct w^{t+1})       (inter-chunk projection)
//   acoef[h][t][n]  = w^{T-1-t}                 (chunk-input accumulation)
//   wT[h][n]        = w^T                       (chunk-to-chunk decay)
__global__ void precompute_kernel(const float* __restrict__ log_dt,
                                  const float* __restrict__ C_re, const float* __restrict__ C_im,
                                  const float* __restrict__ log_A_real,
                                  const float* __restrict__ A_imag,
                                  float* __restrict__ ktap, float* __restrict__ Er,
                                  float* __restrict__ Ei, float2* __restrict__ acoef,
                                  float2* __restrict__ wT) {
  int h = blockIdx.x;
  int n = threadIdx.x;  // 0..31
  float dt  = expf(log_dt[h]);
  float Are = -expf(log_A_real[h * NN + n]);
  float Aim = A_imag[h * NN + n];
  float dr = dt * Are, di = dt * Aim;
  float em = expf(dr);
  float wr = em * cosf(di);
  float wi = em * sinf(di);
  float cr = C_re[h * NN + n], ci = C_im[h * NN + n];
  // Ct = (cr + i ci) * (w - 1) / A
  float nr = cr * (wr - 1.f) - ci * wi;
  float ni = cr * wi + ci * (wr - 1.f);
  float den = Are * Are + Aim * Aim;
  float ctr = (nr * Are + ni * Aim) / den;
  float cti = (ni * Are - nr * Aim) / den;

  float pr = ctr, pi = cti;      // Ct * w^j
  float curr = 1.f, curi = 0.f;  // w^j
  for (int j = 0; j < TT; ++j) {
    float kv = 2.f * pr;
    for (int off = 16; off >= 1; off >>= 1) kv += __shfl_xor(kv, off, 32);
    if (n == 0) ktap[h * TT + j] = kv;
    // advance p -> Ct * w^{j+1}
    float npr = pr * wr - pi * wi;
    float npi = pr * wi + pi * wr;
    pr = npr; pi = npi;
    Er[((size_t)h * TT + j) * NN + n] = 2.f * pr;
    Ei[((size_t)h * TT + j) * NN + n] = 2.f * pi;
    acoef[((size_t)h * TT + (TT - 1 - j)) * NN + n] = make_float2(curr, curi);
    float ncr = curr * wr - curi * wi;
    float nci = curr * wi + curi * wr;
    curr = ncr; curi = nci;
  }
  wT[h * NN + n] = make_float2(curr, curi);  // w^T
}

// ---------------- chunk-boundary state scan (one wave per (b,h)) ----------------
__global__ void carry_kernel(const float* __restrict__ z, const float2* __restrict__ acoef,
                             const float2* __restrict__ wT, float2* __restrict__ Sbuf) {
  int bh = blockIdx.x;  // b*H + h
  int n  = threadIdx.x; // mode
  int h  = bh % HH;
  const float* zrow = z + (size_t)bh * LL;
  __shared__ float zch[TT];
  float2 wt = wT[h * NN + n];
  float sr = 0.f, si = 0.f;
  const float2* ac = acoef + (size_t)h * TT * NN + n;
  for (int c = 0; c < CC; ++c) {
    zch[n]      = zrow[c * TT + n];
    zch[n + 32] = zrow[c * TT + n + 32];
    __syncthreads();
    float ur = 0.f, ui = 0.f;
    for (int t = 0; t < TT; ++t) {
      float zv = zch[t];
      float2 a = ac[(size_t)t * NN];
      ur += a.x * zv;
      ui += a.y * zv;
    }
    float nsr = wt.x * sr - wt.y * si + ur;
    float nsi = wt.x * si + wt.y * sr + ui;
    sr = nsr; si = nsi;
    Sbuf[((size_t)bh * CC + c) * NN + n] = make_float2(sr, si);
    __syncthreads();
  }
}

// ---------------- intra-chunk conv + inter-chunk term + D-skip + GELU ----------------
__global__ void conv_kernel(const float* __restrict__ z, const float2* __restrict__ Sbuf,
                            const float* __restrict__ ktap, const float* __restrict__ Er,
                            const float* __restrict__ Ei, const float* __restrict__ Dp,
                            float* __restrict__ ygelu) {
  int c  = blockIdx.x % CC;
  int bh = blockIdx.x / CC;
  int h  = bh % HH;
  int t  = threadIdx.x;  // 0..63
  __shared__ float zch[TT];
  __shared__ float kch[TT];
  __shared__ float sre[NN];
  __shared__ float sim[NN];
  const float* zrow = z + (size_t)bh * LL + (size_t)c * TT;
  zch[t] = zrow[t];
  kch[t] = ktap[h * TT + t];
  if (t < NN) {
    float2 s = (c == 0) ? make_float2(0.f, 0.f)
                        : Sbuf[((size_t)bh * CC + (c - 1)) * NN + t];
    sre[t] = s.x;
    sim[t] = s.y;
  }
  __syncthreads();
  float y = 0.f;
  for (int m = 0; m <= t; ++m) y += kch[m] * zch[t - m];
  const float* er = Er + ((size_t)h * TT + t) * NN;
  const float* ei = Ei + ((size_t)h * TT + t) * NN;
#pragma unroll
  for (int nn2 = 0; nn2 < NN; ++nn2) y += er[nn2] * sre[nn2] - ei[nn2] * sim[nn2];
  y += Dp[h] * zch[t];
  float g = 0.5f * y * (1.f + erff(y * 0.70710678118f));  // exact GELU
  ygelu[(size_t)bh * LL + (size_t)c * TT + t] = g;
}

// ---------------- channel-mix GEMM (fp32 WMMA) + GLU + residual fused ----------------
// out rows g in [0,256): this block computes both g (top) and g+256 (bottom) so the
// GLU pair lives in the same accumulator coordinates. Block = 8 waves, tile 64M x 128N.
__global__ void glu_gemm_kernel(const float* __restrict__ Y, const float* __restrict__ Wc,
                                const float* __restrict__ bc, const float* __restrict__ hin,
                                float* __restrict__ hpre) {
  int tid  = threadIdx.x;
  int lane = tid & 31;
  int wid  = tid >> 5;
  int wm = wid & 3;   // 4 M-strips of 16
  int wn = wid >> 2;  // 2 N-strips of 64
  int gm = blockIdx.y * 64 + wm * 16;
  int n0 = blockIdx.x * 128 + wn * 64;   // global column (b,l)
  int b  = n0 / LL;
  int l0 = n0 % LL;                      // 128 | 2048, never crosses a batch row
  int lm = lane & 15;
  int lk = (lane >> 4) << 1;             // 0 or 2
  const float* abaseT = Wc + (size_t)(gm + lm) * HH + lk;
  const float* abaseB = Wc + (size_t)(gm + lm + HH) * HH + lk;
  const float* ybase  = Y + ((size_t)b * HH + lk) * LL + l0 + lm;

  v8f zz = {0.f, 0.f, 0.f, 0.f, 0.f, 0.f, 0.f, 0.f};
  v8f accT[4], accB[4];
#pragma unroll
  for (int j = 0; j < 4; ++j) { accT[j] = zz; accB[j] = zz; }

  for (int k = 0; k < HH; k += 4) {
    v2f at = *(const v2f*)(abaseT + k);
    v2f ab = *(const v2f*)(abaseB + k);
    const float* yk = ybase + (size_t)k * LL;
    __builtin_prefetch(yk + 4 * LL, 0, 1);
#pragma unroll
    for (int j = 0; j < 4; ++j) {
      v2f bf;
      bf.x = yk[j * 16];
      bf.y = yk[j * 16 + LL];
      accT[j] = __builtin_amdgcn_wmma_f32_16x16x4_f32(false, at, false, bf,
                                                      (short)0, accT[j], false, false);
      accB[j] = __builtin_amdgcn_wmma_f32_16x16x4_f32(false, ab, false, bf,
                                                      (short)0, accB[j], false, false);
    }
  }

  int colbase = l0 + lm;
#pragma unroll
  for (int r = 0; r < 8; ++r) {
    int g = gm + r + ((lane >= 16) ? 8 : 0);
    float cbt = bc[g];
    float cbb = bc[g + HH];
    size_t rowoff = ((size_t)b * HH + g) * LL;
#pragma unroll
    for (int j = 0; j < 4; ++j) {
      float a  = accT[j][r] + cbt;
      float bv = accB[j][r] + cbb;
      float u  = a / (1.f + expf(-bv));  // a * sigmoid(bv)  (GLU)
      int col = colbase + j * 16;
      u += hin[rowoff + col];            // residual
      hpre[rowoff + col] = u;
    }
  }
}

// ---------------- post-norm LayerNorm over channel dim ----------------
__global__ void ln_kernel(const float* __restrict__ hpre, const float* __restrict__ lw,
                          const float* __restrict__ lb, float* __restrict__ hout) {
  int b  = blockIdx.y;
  int l0 = blockIdx.x * 32;
  int tid = threadIdx.x;
  int lx = tid & 31;   // column within tile (coalesced across lanes)
  int wg = tid >> 5;   // 8 channel groups of 32
  __shared__ float uv[HH][32];  // 32 KB
  __shared__ float ps[8][32];
  __shared__ float pq[8][32];
  __shared__ float mu[32];
  __shared__ float rs[32];
  float s = 0.f, q = 0.f;
  for (int j = 0; j < 32; ++j) {
    int hh = wg * 32 + j;
    float v = hpre[((size_t)b * HH + hh) * LL + l0 + lx];
    uv[hh][lx] = v;
    s += v;
    q += v * v;
  }
  ps[wg][lx] = s;
  pq[wg][lx] = q;
  __syncthreads();
  if (wg == 0) {
    float S = 0.f, Q = 0.f;
    for (int w2 = 0; w2 < 8; ++w2) { S += ps[w2][lx]; Q += pq[w2][lx]; }
    float m = S * (1.f / HH);
    float var = Q * (1.f / HH) - m * m;
    mu[lx] = m;
    rs[lx] = rsqrtf(var + LN_EPS);
  }
  __syncthreads();
  for (int j = 0; j < 32; ++j) {
    int hh = wg * 32 + j;
    float v = (uv[hh][lx] - mu[lx]) * rs[lx] * lw[hh] + lb[hh];
    hout[((size_t)b * HH + hh) * LL + l0 + lx] = v;
  }
}

// ---------------- mean pool over L ----------------
__global__ void pool_kernel(const float* __restrict__ hbuf, float* __restrict__ pooled) {
  int bh = blockIdx.x;
  int tid = threadIdx.x;
  __shared__ float red[256];
  float s = 0.f;
  for (int l = tid; l < LL; l += 256) s += hbuf[(size_t)bh * LL + l];
  red[tid] = s;
  __syncthreads();
  for (int off = 128; off > 0; off >>= 1) {
    if (tid < off) red[tid] += red[tid + off];
    __syncthreads();
  }
  if (tid == 0) pooled[bh] = red[0] * (1.f / LL);
}

// ---------------- decoder ----------------
__global__ void dec_kernel(const float* __restrict__ pooled, const float* __restrict__ dw,
                           const float* __restrict__ db, float* __restrict__ out) {
  int b = blockIdx.x;
  int d = threadIdx.x;
  if (d >= DOUTN) return;
  float acc = db[d];
  for (int hh = 0; hh < HH; ++hh) acc += pooled[b * HH + hh] * dw[d * HH + hh];
  out[b * DOUTN + d] = acc;
}

extern "C" void kernel_launch(void* const* d_in, const int* in_sizes, int n_in,
                              void* d_out, int out_size, void* d_ws, size_t ws_size,
                              hipStream_t stream) {
  (void)in_sizes; (void)n_in; (void)out_size; (void)ws_size;
  const float* x          = (const float*)d_in[0];
  const float* enc_w      = (const float*)d_in[1];
  const float* enc_b      = (const float*)d_in[2];
  const float* log_dt     = (const float*)d_in[3];
  const float* C_re       = (const float*)d_in[4];
  const float* C_im       = (const float*)d_in[5];
  const float* log_A_real = (const float*)d_in[6];
  const float* A_imag     = (const float*)d_in[7];
  const float* Dp         = (const float*)d_in[8];
  const float* conv_w     = (const float*)d_in[9];
  const float* conv_b     = (const float*)d_in[10];
  const float* ln_w       = (const float*)d_in[11];
  const float* ln_b       = (const float*)d_in[12];
  const float* dec_w      = (const float*)d_in[13];
  const float* dec_b      = (const float*)d_in[14];
  float* out = (float*)d_out;

  char* ws = (char*)d_ws;
  size_t szH = (size_t)BB * HH * LL * sizeof(float);  // 64 MB
  float*  hbuf  = (float*)(ws);
  float*  ygelu = (float*)(ws + szH);
  float2* Sbuf  = (float2*)(ws + 2 * szH);  // 64 MB (B*H*C*N2 complex)
  float*  hpre  = (float*)(ws + 2 * szH);   // aliases Sbuf (dead by then)
  char* cbase = ws + 3 * szH;
  float* ktap = (float*)cbase;  cbase += (size_t)HH * TT * sizeof(float);
  float* Er   = (float*)cbase;  cbase += (size_t)HH * TT * NN * sizeof(float);
  float* Ei   = (float*)cbase;  cbase += (size_t)HH * TT * NN * sizeof(float);
  float2* acoef = (float2*)cbase; cbase += (size_t)HH * TT * NN * sizeof(float2);
  float2* wTp   = (float2*)cbase; cbase += (size_t)HH * NN * sizeof(float2);
  float* pooled = (float*)cbase;

  {
    size_t tot = (size_t)BB * HH * LL;
    int blocks = (int)((tot + 255) / 256);
    enc_kernel<<<blocks, 256, 0, stream>>>(x, enc_w, enc_b, hbuf);
  }
  for (int i = 0; i < NLAYER; ++i) {
    precompute_kernel<<<HH, NN, 0, stream>>>(
        log_dt + (size_t)i * HH, C_re + (size_t)i * HH * NN, C_im + (size_t)i * HH * NN,
        log_A_real + (size_t)i * HH * NN, A_imag + (size_t)i * HH * NN,
        ktap, Er, Ei, acoef, wTp);
    carry_kernel<<<BB * HH, NN, 0, stream>>>(hbuf, acoef, wTp, Sbuf);
    conv_kernel<<<BB * HH * CC, TT, 0, stream>>>(hbuf, Sbuf, ktap, Er, Ei,
                                                 Dp + (size_t)i * HH, ygelu);
    dim3 gg(BB * LL / 128, HH / 64);
    glu_gemm_kernel<<<gg, 256, 0, stream>>>(ygelu, conv_w + (size_t)i * 2 * HH * HH,
                                            conv_b + (size_t)i * 2 * HH, hbuf, hpre);
    dim3 gl(LL / 32, BB);
    ln_kernel<<<gl, 256, 0, stream>>>(hpre, ln_w + (size_t)i * HH, ln_b + (size_t)i * HH, hbuf);
  }
  pool_kernel<<<BB * HH, 256, 0, stream>>>(hbuf, pooled);
  dec_kernel<<<BB, 256, 0, stream>>>(pooled, dec_w, dec_b, out);
}